// HybridSharpModel_60936995995891
// MI455X (gfx1250) — compile-verified
//
#include <hip/hip_runtime.h>

typedef __attribute__((ext_vector_type(2))) float v2f;
typedef __attribute__((ext_vector_type(8))) float v8f;

#define PI_F 3.14159265358979323846f

// ---------------------------------------------------------------------------
// Pre-kernel: compute spectrally-normalized S (5x5) and cos/sin(raw_phase)
// into workspace:  ws[0..24]=S, ws[25..49]=cos(raw_phase), ws[50..74]=sin.
// ---------------------------------------------------------------------------
__global__ void hsm_prep(const float* __restrict__ raw_S,
                         const float* __restrict__ raw_phase,
                         const float* __restrict__ u,
                         const int*   __restrict__ epoch,
                         float* __restrict__ ws)
{
    __shared__ float Score[25];
    __shared__ float factor_sh;
    const int t = threadIdx.x;
    const float tau = fminf(0.15f, (float)epoch[0] * (1.0f / 15000.0f));
    if (t < 25) {
        const int i = t / 5, j = t - 5 * (t / 5);
        const float rs = raw_S[t];
        const float S  = tanhf(rs * 3.0f);
        const float mag = fmaxf(fabsf(S) - tau, 0.0f);
        const float ssp = (S > 0.0f) ? mag : ((S < 0.0f) ? -mag : 0.0f);
        Score[t] = (i == j) ? (0.95f + 0.04f / (1.0f + expf(-rs))) : ssp;
    }
    __syncthreads();
    if (t == 0) {
        // sigma = || u @ S_core ||  (since v = uS/||uS||, uS.v = ||uS||)
        float sig2 = 0.0f;
        for (int j = 0; j < 5; ++j) {
            float uS = 0.0f;
            for (int i = 0; i < 5; ++i) uS += u[i] * Score[i * 5 + j];
            sig2 += uS * uS;
        }
        const float sigma = sqrtf(sig2);
        factor_sh = 1.05f / fmaxf(sigma, 1.05f);
    }
    __syncthreads();
    if (t < 25) {
        ws[t] = Score[t] * factor_sh;
        float s, c;
        sincosf(raw_phase[t], &s, &c);
        ws[25 + t] = c;
        ws[50 + t] = s;
    }
}

// ---------------------------------------------------------------------------
// Main fused kernel (persistent blocks): 128 threads = 4 waves, each wave
// owns 16 batch rows per chunk; blocks grid-stride over 64-row chunks so the
// invariant LDS staging (W1 transpose, LN/GEMM2 constants, S tables) is paid
// once per block instead of once per 64 rows.
//   GEMM1 (Bx25 @ 25x128) via V_WMMA_F32_16X16X4_F32  (K padded 25->28)
//   W1 staged as K-pairs so each B fragment is ONE aligned ds_load_b64
//   LayerNorm + ReLU + GEMM2 (128->4) via VALU (N=4 too thin for WMMA)
//   Per-row rotation/contraction with angle-addition identities.
// ---------------------------------------------------------------------------
__global__ __launch_bounds__(128) void hsm_main(
    const float* __restrict__ x_2ch,
    const float* __restrict__ history,
    const float* __restrict__ W1,
    const float* __restrict__ b1,
    const float* __restrict__ gamma,
    const float* __restrict__ beta,
    const float* __restrict__ W2,
    const float* __restrict__ b2,
    const float* __restrict__ ws,
    float* __restrict__ out,
    int nChunks,
    int Bsz)
{
    // W1 transposed, stored as K-pairs: Bpair[k>>1][n][k&1] = W1[n][k]
    // -> B fragment (K=k,k+1 at column n) is one contiguous 8B-aligned load.
    __shared__ float Bpair[14][128][2];
    __shared__ float Ahist[4][16][28];   // per-wave history tile, zero-padded
    __shared__ float Hbuf[4][16][133];   // per-wave h tile (pad 133: stride%64=5)
    __shared__ float Ssh[25], CPsh[25], SPsh[25];
    __shared__ float B1sh[128], Gsh[128], Besh[128], W2sh[512], B2sh[4];
    __shared__ float StatSh[4][2][16][2];
    __shared__ float Pacc[4][2][16][4];

    const int tid  = threadIdx.x;
    const int wave = tid >> 5;
    const int lane = tid & 31;

    // ---- stage invariant constants into LDS (once per persistent block) ----
    {
        const int d = tid;
        for (int k = 0; k < 25; ++k) Bpair[k >> 1][d][k & 1] = W1[d * 25 + k];
        Bpair[12][d][1] = 0.0f;          // k = 25
        Bpair[13][d][0] = 0.0f;          // k = 26
        Bpair[13][d][1] = 0.0f;          // k = 27
        B1sh[d] = b1[d]; Gsh[d] = gamma[d]; Besh[d] = beta[d];
        #pragma unroll
        for (int p = 0; p < 4; ++p) W2sh[p * 128 + d] = W2[p * 128 + d];
        if (d < 4)  B2sh[d] = b2[d];
        if (d < 25) { Ssh[d] = ws[d]; CPsh[d] = ws[25 + d]; SPsh[d] = ws[50 + d]; }
    }

    const int m16  = lane & 15;
    const int half = lane >> 4;
    const int koff = half << 1;                 // A/B frag K pair base: 0 or 2
    const int mrow = half << 3;                 // C rows: 0..7 or 8..15
    const int row  = lane & 15;
    const int j0   = half * 64;

    for (int chunk = blockIdx.x; chunk < nChunks; chunk += gridDim.x) {
        const long long rowBase = (long long)chunk * 64 + (long long)wave * 16;

        // ---- prefetch next chunk's history tile toward caches ----
        {
            const int nextChunk = chunk + gridDim.x;
            if (nextChunk < nChunks && lane < 13) {
                const char* p = (const char*)(history +
                    ((long long)nextChunk * 64 + (long long)wave * 16) * 25)
                    + (size_t)lane * 128;
                __builtin_prefetch(p, 0, 3);    // global_prefetch_b8
            }
        }

        // ---- load 16x25 history tile (coalesced) + zero pad K=25..27 ----
        {
            const float* hsrc = history + rowBase * 25;
            float* adst = &Ahist[wave][0][0];
            for (int idx = lane; idx < 400; idx += 32) {
                const int m = idx / 25, j = idx - m * 25;
                adst[m * 28 + j] = hsrc[idx];
            }
            for (int idx = lane; idx < 48; idx += 32) {
                const int m = idx / 3, j = 25 + (idx - m * 3);
                adst[m * 28 + j] = 0.0f;
            }
        }
        __syncthreads();

        // ---- GEMM1: h = history @ W1^T + b1 via f32 WMMA 16x16x4 ----
        // A frag: lanes 0-15 hold (K=0,1), lanes 16-31 hold (K=2,3), M=lane&15.
        // Row stride 28 floats (112 B) and even K offset -> 8B-aligned b64 loads.
        v2f af[7];
        #pragma unroll
        for (int kk = 0; kk < 7; ++kk)
            af[kk] = *(const v2f*)&Ahist[wave][m16][4 * kk + koff];

        #pragma unroll
        for (int t = 0; t < 8; ++t) {
            const float binit = B1sh[16 * t + m16]; // C col = 16t+m16 for all 8 regs
            v8f c = {binit, binit, binit, binit, binit, binit, binit, binit};
            #pragma unroll
            for (int kk = 0; kk < 7; ++kk) {
                // B frag: K pair (4kk+koff, +1) at column 16t+m16 -> one b64 load
                const v2f bf = *(const v2f*)&Bpair[2 * kk + half][16 * t + m16][0];
                c = __builtin_amdgcn_wmma_f32_16x16x4_f32(false, af[kk], false, bf,
                                                          (short)0, c, false, false);
            }
            #pragma unroll
            for (int r = 0; r < 8; ++r)
                Hbuf[wave][mrow + r][16 * t + m16] = c[r];
        }
        __syncthreads();

        // ---- LayerNorm stats: each half-wave sums 64 cols of its row ----
        {
            float sum = 0.0f, sq = 0.0f;
            for (int j = j0; j < j0 + 64; ++j) {
                const float v = Hbuf[wave][row][j];
                sum += v; sq = fmaf(v, v, sq);
            }
            StatSh[wave][half][row][0] = sum;
            StatSh[wave][half][row][1] = sq;
        }
        __syncthreads();
        float mu, rstd;
        {
            const float sum = StatSh[wave][0][row][0] + StatSh[wave][1][row][0];
            const float sq  = StatSh[wave][0][row][1] + StatSh[wave][1][row][1];
            mu = sum * (1.0f / 128.0f);
            const float var = sq * (1.0f / 128.0f) - mu * mu;
            rstd = rsqrtf(var + 1e-5f);
        }

        // ---- GEMM2 partials: relu(LN(h)) @ W2^T (4 outs, split over halves) ----
        float a0 = 0.0f, a1 = 0.0f, a2 = 0.0f, a3 = 0.0f;
        for (int j = j0; j < j0 + 64; ++j) {
            float v = (Hbuf[wave][row][j] - mu) * rstd * Gsh[j] + Besh[j];
            v = fmaxf(v, 0.0f);
            a0 = fmaf(v, W2sh[j],       a0);
            a1 = fmaf(v, W2sh[128 + j], a1);
            a2 = fmaf(v, W2sh[256 + j], a2);
            a3 = fmaf(v, W2sh[384 + j], a3);
        }
        Pacc[wave][half][row][0] = a0; Pacc[wave][half][row][1] = a1;
        Pacc[wave][half][row][2] = a2; Pacc[wave][half][row][3] = a3;
        __syncthreads();

        // ---- finalize one batch row per lane (lanes 0-15 of each wave) ----
        if (half == 0) {
            const float p0 = Pacc[wave][0][row][0] + Pacc[wave][1][row][0] + B2sh[0];
            const float p1 = Pacc[wave][0][row][1] + Pacc[wave][1][row][1] + B2sh[1];
            const float p2 = Pacc[wave][0][row][2] + Pacc[wave][1][row][2] + B2sh[2];
            const float p3 = Pacc[wave][0][row][3] + Pacc[wave][1][row][3] + B2sh[3];

            const float phi0 = tanhf(p0) * (PI_F / 36.0f);
            const float phi1 = 0.5f * PI_F + tanhf(p2) * (PI_F / 36.0f);
            const float r0 = 1.2f / (1.0f + expf(-p1));
            const float r1 = 0.6f / (1.0f + expf(-p3));
            float c0, s0, c1, s1;
            sincosf(phi0, &s0, &c0);
            sincosf(phi1, &s1, &c1);

            const long long b = rowBase + row;
            const float* xp = x_2ch + b * 10;
            float xr[5], xi[5];
            #pragma unroll
            for (int n = 0; n < 5; ++n) { xr[n] = xp[2 * n]; xi[n] = xp[2 * n + 1]; }

            float* outp = out + b * 10;
            #pragma unroll
            for (int i = 0; i < 5; ++i) {
                float sr = 0.0f, si = 0.0f;
                #pragma unroll
                for (int j = 0; j < 5; ++j) {
                    const float cp = CPsh[i * 5 + j], sp = SPsh[i * 5 + j];
                    const float Sij = Ssh[i * 5 + j];
                    if (j == i) {   // mode 1 (eye), rotation by phi1
                        const float C  = cp * c1 - sp * s1;
                        const float Sn = sp * c1 + cp * s1;
                        const float a  = r1 * Sij;
                        sr = fmaf(a, C * xr[j] - Sn * xi[j], sr);
                        si = fmaf(a, Sn * xr[j] + C * xi[j], si);
                    } else {        // mode 0 (1-eye), rotation by phi0
                        const float C  = cp * c0 - sp * s0;
                        const float Sn = sp * c0 + cp * s0;
                        const float a  = r0 * Sij;
                        sr = fmaf(a, C * xr[j] - Sn * xi[j], sr);
                        si = fmaf(a, Sn * xr[j] + C * xi[j], si);
                    }
                }
                outp[2 * i]     = 0.999f * sr;
                outp[2 * i + 1] = 0.999f * si;
            }
            float* phip = out + (long long)Bsz * 10 + b * 2;
            phip[0] = phi0;
            phip[1] = phi1;
        }
        __syncthreads();   // phase buffers recycled next chunk iteration
    }
}

// ---------------------------------------------------------------------------
extern "C" void kernel_launch(void* const* d_in, const int* in_sizes, int n_in,
                              void* d_out, int out_size, void* d_ws, size_t ws_size,
                              hipStream_t stream)
{
    const float* x_2ch     = (const float*)d_in[0];
    const float* history   = (const float*)d_in[1];
    // d_in[2] = phi_global (unused by the reference)
    const float* raw_S     = (const float*)d_in[3];
    const float* raw_phase = (const float*)d_in[4];
    const float* W1        = (const float*)d_in[5];
    const float* b1        = (const float*)d_in[6];
    const float* gamma     = (const float*)d_in[7];
    const float* beta      = (const float*)d_in[8];
    const float* W2        = (const float*)d_in[9];
    const float* b2        = (const float*)d_in[10];
    const float* u         = (const float*)d_in[11];
    const int*   epoch     = (const int*)d_in[12];

    float* out = (float*)d_out;
    float* ws  = (float*)d_ws;

    const int Bsz = in_sizes[0] / 10;      // x_2ch has B*5*2 elements

    hsm_prep<<<1, 32, 0, stream>>>(raw_S, raw_phase, u, epoch, ws);

    const int nChunks = Bsz / 64;          // 4 waves * 16 rows per chunk
    int blocks = nChunks < 2048 ? nChunks : 2048;   // persistent blocks
    hsm_main<<<blocks, 128, 0, stream>>>(x_2ch, history, W1, b1, gamma, beta,
                                         W2, b2, ws, out, nChunks, Bsz);
}